// PointsRenderer_13486197309906
// MI455X (gfx1250) — compile-verified
//
#include <hip/hip_runtime.h>
#include <stdint.h>

// Problem constants (from reference): B=4, N=100000, F=64, S=256, K=8
constexpr int   KK     = 8;
constexpr int   FF     = 64;
constexpr int   SSQ    = 256 * 256;     // S*S
constexpr int   PIX    = 256;           // pixels (threads) per block
constexpr float INV_R2 = 16384.0f;      // 1/(R*R), R = 2*RADIUS/S = 2/256

// Async global->LDS 16B copy (CDNA5 async engine, tracked by ASYNCcnt).
// Non-temporal: metadata is read exactly once -- don't evict the L2-resident
// feature table with it.
// %0 = LDS byte offset (VGPR), %1 = 64-bit global address (VGPR pair).
__device__ __forceinline__ void async_cp16_nt(uint32_t lds_off, const void* gaddr) {
    asm volatile("global_load_async_to_lds_b128 %0, %1, off th:TH_LOAD_NT"
                 :: "v"(lds_off), "v"(gaddr)
                 : "memory");
}

__global__ __launch_bounds__(PIX, 1)
void points_renderer_kernel(const float* __restrict__ feats,      // (B*N, 64)
                            const long long* __restrict__ idx,    // (B*S*S, 8)
                            const float* __restrict__ zbuf,       // (B*S*S, 8)
                            const float* __restrict__ dist,       // (B*S*S, 8)
                            float* __restrict__ out,              // (B, 65, S, S)
                            int total_points)                     // B*N
{
    __shared__ __align__(16) long long s_idx[PIX * KK];   // 16 KB
    __shared__ __align__(16) float     s_z[PIX * KK];     //  8 KB
    __shared__ __align__(16) float     s_d[PIX * KK];     //  8 KB

    const int tid = threadIdx.x;
    const int p   = blockIdx.x * PIX + tid;               // global pixel id
    const size_t rowLL = (size_t)blockIdx.x * PIX * KK;

    // ---- Stage per-pixel metadata through LDS via the async engine --------
    {
        const char* gi = (const char*)(idx  + rowLL) + tid * 64;
        const char* gz = (const char*)(zbuf + rowLL) + tid * 32;
        const char* gd = (const char*)(dist + rowLL) + tid * 32;
        const uint32_t li = (uint32_t)(uintptr_t)(const void*)&s_idx[tid * KK];
        const uint32_t lz = (uint32_t)(uintptr_t)(const void*)&s_z[tid * KK];
        const uint32_t ld = (uint32_t)(uintptr_t)(const void*)&s_d[tid * KK];
        async_cp16_nt(li +  0, gi +  0);
        async_cp16_nt(li + 16, gi + 16);
        async_cp16_nt(li + 32, gi + 32);
        async_cp16_nt(li + 48, gi + 48);
        async_cp16_nt(lz +  0, gz +  0);
        async_cp16_nt(lz + 16, gz + 16);
        async_cp16_nt(ld +  0, gd +  0);
        async_cp16_nt(ld + 16, gd + 16);
    }
    asm volatile("s_wait_asynccnt 0" ::: "memory");
    // Each lane consumes only the slice it staged itself -> no barrier needed.

    // ---- Weights / transmittance / depth ---------------------------------
    float w[KK], zv[KK], contrib[KK];
    int   sp[KK];
    #pragma unroll
    for (int k = 0; k < KK; ++k) {
        const long long pi = s_idx[tid * KK + k];
        float dd = fmaxf(s_d[tid * KK + k] * INV_R2, 0.0f);
        float a  = fminf(expf(-dd), 0.99f);
        w[k]  = (pi >= 0) ? a : 0.0f;
        zv[k] = s_z[tid * KK + k];
        long long c = (pi < 0) ? 0 : pi;
        if (c > (long long)(total_points - 1)) c = total_points - 1;
        sp[k] = (int)c;
    }

    float trans = 1.0f, wsum = 0.0f, zacc = 0.0f;
    #pragma unroll
    for (int k = 0; k < KK; ++k) {
        contrib[k] = w[k] * trans;
        trans *= (1.0f - w[k]);
        wsum  += w[k];
        zacc  += zv[k] * w[k];
    }
    const float depth = zacc / fmaxf(wsum, 1e-9f);

    // ---- Prefetch the 8 feature rows (256B each); default RT policy so the
    //      feature table stays L2-resident (102 MB table < 192 MB L2) -------
    #pragma unroll
    for (int k = 0; k < KK; ++k) {
        const char* pr = (const char*)(feats + (size_t)sp[k] * FF);
        __builtin_prefetch(pr,       0, 0);
        __builtin_prefetch(pr + 128, 0, 0);
    }

    // ---- Gather + composite: each lane streams whole 256B rows (B128) ----
    float acc[FF];
    #pragma unroll
    for (int f = 0; f < FF; ++f) acc[f] = 0.0f;

    #pragma unroll 1   // keep VGPRs bounded; 16 B128s in flight per iter
    for (int k = 0; k < KK; ++k) {
        const float4* __restrict__ src =
            (const float4*)(feats + (size_t)sp[k] * FF);
        const float c = contrib[k];
        #pragma unroll
        for (int j = 0; j < FF / 4; ++j) {
            const float4 g = src[j];
            acc[4*j+0] = fmaf(c, g.x, acc[4*j+0]);
            acc[4*j+1] = fmaf(c, g.y, acc[4*j+1]);
            acc[4*j+2] = fmaf(c, g.z, acc[4*j+2]);
            acc[4*j+3] = fmaf(c, g.w, acc[4*j+3]);
        }
    }

    // ---- Coalesced non-temporal stores (write-once output; don't pollute
    //      L2 that holds the feature table) --------------------------------
    const int b  = p / SSQ;
    const int yx = p - b * SSQ;
    float* ob = out + (size_t)b * 65 * SSQ + yx;
    #pragma unroll
    for (int f = 0; f < FF; ++f)
        __builtin_nontemporal_store(acc[f], &ob[(size_t)f * SSQ]);
    __builtin_nontemporal_store(depth, &ob[(size_t)64 * SSQ]);
}

extern "C" void kernel_launch(void* const* d_in, const int* in_sizes, int n_in,
                              void* d_out, int out_size, void* d_ws, size_t ws_size,
                              hipStream_t stream) {
    const float*     feats = (const float*)d_in[0];      // (B, N, 64) f32
    const long long* idx   = (const long long*)d_in[1];  // (B, S, S, 8) i64
    const float*     zbuf  = (const float*)d_in[2];      // (B, S, S, 8) f32
    const float*     dist  = (const float*)d_in[3];      // (B, S, S, 8) f32
    float*           out   = (float*)d_out;              // (B, 65, S, S) f32

    const int total_points = in_sizes[0] / FF;   // B*N
    const int n_pixels     = in_sizes[2] / KK;   // B*S*S
    const int n_blocks     = n_pixels / PIX;     // 1024

    points_renderer_kernel<<<dim3(n_blocks), dim3(PIX), 0, stream>>>(
        feats, idx, zbuf, dist, out, total_points);
}